// TransOp_expm_85985245266203
// MI455X (gfx1250) — compile-verified
//
#include <hip/hip_runtime.h>

typedef __attribute__((ext_vector_type(2))) float v2f;
typedef __attribute__((ext_vector_type(4))) float v4f;
typedef __attribute__((ext_vector_type(8))) float v8f;

#define ORD 8   // Taylor/Horner order
#define SQS 6   // squarings (scale = 2^-6)

// One block = 16 waves (512 threads) = 16 batch elements.
// Phase 1: T_tile[16x256] = C_tile[16x64] x Psi[64x256] on v_wmma_f32_16x16x4_f32.
// Phase 2: per-wave expm via fully-unrolled Horner + squaring, all matmuls on
//          WMMA, layout conversion through a wave-private padded LDS tile.
__global__ __launch_bounds__(512) void expm_apply_kernel(
    const float* __restrict__ x,    // [B,16]
    const float* __restrict__ c,    // [B,64]
    const float* __restrict__ psi,  // [64,16,16]
    float* __restrict__ out)        // [B,16]
{
    // 16 tiles of 16x16 f32, row stride 17 floats to avoid bank conflicts
    __shared__ float tile[16][16 * 17];

    const int tid = threadIdx.x;
    const int w   = tid >> 5;   // wave id 0..15
    const int l   = tid & 31;   // lane 0..31
    const int h   = l >> 4;     // half of wave
    const int n   = l & 15;     // 0..15
    const long b0 = (long)blockIdx.x * 16;

    // ---------------- Phase 1: build T for 16 batch elements via WMMA ----
    // Wave w computes D[b', n] = sum_m c[b0+b', m] * psi[m, w, n]
    // A layout (16x4 f32): lanes 0-15 -> M rows, VGPR0/1 = K, K+1; lanes 16-31 K+2,K+3
    v8f acc = {0.f, 0.f, 0.f, 0.f, 0.f, 0.f, 0.f, 0.f};
#pragma unroll
    for (int kk = 0; kk < 16; ++kk) {
        const int k0 = kk * 4;
        v2f a;
        const float* cp = c + (b0 + n) * 64 + (k0 + 2 * h);
        a.x = cp[0];
        a.y = cp[1];
        // B layout (4x16 f32): VGPR0 rows K (lanes0-15) / K+2 (lanes16-31); VGPR1 +1
        const float* pp = psi + (size_t)(k0 + 2 * h) * 256 + w * 16 + n;
        v2f bb;
        bb.x = pp[0];
        bb.y = pp[256];
        acc = __builtin_amdgcn_wmma_f32_16x16x4_f32(
            false, a, false, bb, (short)0, acc, false, false);
    }
    // Scatter: acc[j] = T_{b'=j+8h}[row=w][col=n]
#pragma unroll
    for (int j = 0; j < 8; ++j) {
        tile[j + 8 * h][w * 17 + n] = acc[j];
    }
    __syncthreads();

    // ---------------- Phase 2: expm per wave -----------------------------
    float* mt = &tile[w][0];

    // Fixed A-layout registers of scaled matrix A = T * 2^-SQS
    const float sc = 1.0f / (float)(1 << SQS);
    v2f Ak[4];
#pragma unroll
    for (int kk = 0; kk < 4; ++kk) {
        const int col = kk * 4 + 2 * h;
        Ak[kk].x = mt[n * 17 + col] * sc;
        Ak[kk].y = mt[n * 17 + col + 1] * sc;
    }

    // R = I in C/D layout: lane holds rows M=j+8h, col N=n
    v8f r;
#pragma unroll
    for (int j = 0; j < 8; ++j) r[j] = (n == (j + 8 * h)) ? 1.f : 0.f;

    // Horner: for k = ORD..1: R = I + (A*R)/k  (fully unrolled; invk literal)
#pragma unroll
    for (int k = ORD; k >= 1; --k) {
#pragma unroll
        for (int j = 0; j < 8; ++j) mt[(j + 8 * h) * 17 + n] = r[j];
        asm volatile("s_wait_dscnt 0" ::: "memory");
        v8f d = {0.f, 0.f, 0.f, 0.f, 0.f, 0.f, 0.f, 0.f};
#pragma unroll
        for (int kk = 0; kk < 4; ++kk) {
            const int row = kk * 4 + 2 * h;
            v2f bb;
            bb.x = mt[row * 17 + n];
            bb.y = mt[(row + 1) * 17 + n];
            d = __builtin_amdgcn_wmma_f32_16x16x4_f32(
                false, Ak[kk], false, bb, (short)0, d, false, false);
        }
        const float invk = 1.0f / (float)k;  // compile-time constant after unroll
#pragma unroll
        for (int j = 0; j < 8; ++j)
            r[j] = ((n == (j + 8 * h)) ? 1.f : 0.f) + d[j] * invk;
    }

    // Squarings: R = R*R, SQS times (reload both A and B layouts from LDS)
#pragma unroll
    for (int sq = 0; sq < SQS; ++sq) {
#pragma unroll
        for (int j = 0; j < 8; ++j) mt[(j + 8 * h) * 17 + n] = r[j];
        asm volatile("s_wait_dscnt 0" ::: "memory");
        v8f d = {0.f, 0.f, 0.f, 0.f, 0.f, 0.f, 0.f, 0.f};
#pragma unroll
        for (int kk = 0; kk < 4; ++kk) {
            const int col = kk * 4 + 2 * h;
            v2f aa;
            aa.x = mt[n * 17 + col];
            aa.y = mt[n * 17 + col + 1];
            v2f bb;
            bb.x = mt[col * 17 + n];
            bb.y = mt[(col + 1) * 17 + n];
            d = __builtin_amdgcn_wmma_f32_16x16x4_f32(
                false, aa, false, bb, (short)0, d, false, false);
        }
#pragma unroll
        for (int j = 0; j < 8; ++j) r[j] = d[j];
    }

    // ---------------- out[b] = expT @ x[b] --------------------------------
    const long b = b0 + w;
    const float xv = x[b * 16 + n];
    float p[8];
#pragma unroll
    for (int j = 0; j < 8; ++j) p[j] = r[j] * xv;
#pragma unroll
    for (int off = 1; off < 16; off <<= 1) {
#pragma unroll
        for (int j = 0; j < 8; ++j) p[j] += __shfl_xor(p[j], off, 16);
    }
    if (n == 0) {
        // two contiguous float4 stores -> global_store_b128
        v4f lo = {p[0], p[1], p[2], p[3]};
        v4f hi = {p[4], p[5], p[6], p[7]};
        float* op = out + b * 16 + 8 * h;
        *(v4f*)(op)     = lo;
        *(v4f*)(op + 4) = hi;
    }
}

extern "C" void kernel_launch(void* const* d_in, const int* in_sizes, int n_in,
                              void* d_out, int out_size, void* d_ws, size_t ws_size,
                              hipStream_t stream) {
    const float* x   = (const float*)d_in[0];  // [B,16,1]
    const float* c   = (const float*)d_in[1];  // [B,64]
    const float* psi = (const float*)d_in[2];  // [64,16,16]
    float* out = (float*)d_out;                // [B,16,1]

    const int B = in_sizes[1] / 64;            // 65536
    dim3 grid(B / 16);
    dim3 block(512);
    hipLaunchKernelGGL(expm_apply_kernel, grid, block, 0, stream,
                       x, c, psi, out);
}